// WGraphSAGE_4776003633677
// MI455X (gfx1250) — compile-verified
//
#include <hip/hip_runtime.h>

namespace {

constexpr int NN = 40000;   // nodes
constexpr int NE = 640000;  // edges
constexpr int FF = 128;     // F_IN == H
constexpr int CC = 32;      // classes

typedef __attribute__((ext_vector_type(2))) float v2f;
typedef __attribute__((ext_vector_type(8))) float v8f;

__global__ void fill_zero_kernel(float* __restrict__ p, int n) {
  int i = blockIdx.x * blockDim.x + threadIdx.x;
  if (i < n) p[i] = 0.0f;
}

__global__ void edge_stats_kernel(const int* __restrict__ dst,
                                  const float* __restrict__ w,
                                  float* __restrict__ deg,
                                  float* __restrict__ sumw) {
  int e = blockIdx.x * blockDim.x + threadIdx.x;
  if (e >= NE) return;
  int d = dst[e];
  atomicAdd(&deg[d], 1.0f);
  atomicAdd(&sumw[d], w[e]);
}

__global__ void node_coef_kernel(const float* __restrict__ deg,
                                 const float* __restrict__ sumw,
                                 float* __restrict__ alpha,
                                 float* __restrict__ beta) {
  int n = blockIdx.x * blockDim.x + threadIdx.x;
  if (n >= NN) return;
  float dg = deg[n];
  float a = 0.0f, b = 0.0f;
  if (dg > 0.0f) {
    float sw = sumw[n];
    if (sw == 0.0f) sw = 1.0f;
    a = dg / ((dg + 1.0f) * sw);
    b = 1.0f / (dg + 1.0f);
  }
  alpha[n] = a;
  beta[n] = b;
}

// One wave32 per edge: lane l handles features [4l, 4l+4).
__global__ void scatter_kernel(const float* __restrict__ h,
                               const int* __restrict__ src,
                               const int* __restrict__ dst,
                               const float* __restrict__ w,
                               float* __restrict__ summ) {
  unsigned gtid = blockIdx.x * blockDim.x + threadIdx.x;
  unsigned e = gtid >> 5;
  unsigned lane = gtid & 31u;
  if (e >= (unsigned)NE) return;
  int s = src[e];
  int d = dst[e];
  float wv = w[e];
  float4 v = ((const float4*)(h + (size_t)s * FF))[lane];
  float* out = summ + (size_t)d * FF + lane * 4;
  atomicAdd(out + 0, v.x * wv);
  atomicAdd(out + 1, v.y * wv);
  atomicAdd(out + 2, v.z * wv);
  atomicAdd(out + 3, v.w * wv);
}

// Y[NN][NCOLS] = act( A_eff[NN][FF] @ Wm[NCOLS][FF]^T + bias )
// A_eff = FUSE ? (alpha[n]*sum_m + beta[n]*h) : A0   (fused neigh computation)
// One wave per 16x16 tile, K consumed by chained v_wmma_f32_16x16x4_f32.
template <int NCOLS, bool FUSE, bool RELU, int TM, int TN>
__global__ __launch_bounds__(256) void gemm_wmma_kernel(
    const float* __restrict__ A0,     // FUSE ? sum_m : X
    const float* __restrict__ Hprev,  // FUSE ? h_prev : unused
    const float* __restrict__ alpha,
    const float* __restrict__ beta,
    const float* __restrict__ Wm,     // [NCOLS][FF] row-major
    const float* __restrict__ bias,   // [NCOLS]
    float* __restrict__ Y)            // [NN][NCOLS]
{
  static_assert(TM * TN == 8, "8 waves per 256-thread block");
  constexpr int LDSW = FF + 4;  // pad: bank = 4n mod 64, conflict-free
  __shared__ float w_lds[NCOLS * LDSW];

  // Cooperative stage of W into LDS (whole matrix, reused by all 8 waves).
  for (int idx = threadIdx.x; idx < NCOLS * FF; idx += 256) {
    int r = idx / FF, c = idx % FF;
    w_lds[r * LDSW + c] = Wm[idx];
  }
  __syncthreads();

  const int wave = threadIdx.x >> 5;
  const int lane = threadIdx.x & 31;
  const int mtile = blockIdx.x * TM + wave / TN;
  const int row0 = mtile * 16;
  const int col0 = (wave % TN) * 16;

  // ISA A-frag layout (16x4 f32): lane L -> M = L%16; VGPR0/1 hold
  // K = khalf+0 / khalf+1 with khalf = (L>=16) ? 2 : 0. B mirrored on N.
  const int m = lane & 15;
  const int khalf = (lane >> 4) << 1;
  const int row = row0 + m;
  const int ncol = col0 + m;

  float al = 0.0f, be = 0.0f;
  if constexpr (FUSE) {
    al = alpha[row];
    be = beta[row];
  }

  const float* arow = A0 + (size_t)row * FF + khalf;
  const float* hrow = FUSE ? (Hprev + (size_t)row * FF + khalf) : arow;
  const float* brow = &w_lds[ncol * LDSW + khalf];

  v8f acc = {};
#pragma unroll
  for (int k0 = 0; k0 < FF; k0 += 4) {
    v2f a, b;
    float2 av = *(const float2*)(arow + k0);
    if constexpr (FUSE) {
      float2 hv = *(const float2*)(hrow + k0);
      a.x = al * av.x + be * hv.x;  // neigh = alpha*sum_m + beta*h (fused)
      a.y = al * av.y + be * hv.y;
    } else {
      a.x = av.x;
      a.y = av.y;
    }
    float2 bv = *(const float2*)(brow + k0);
    b.x = bv.x;
    b.y = bv.y;
    acc = __builtin_amdgcn_wmma_f32_16x16x4_f32(false, a, false, b,
                                                (short)0, acc, false, false);
  }

  // C/D layout: VGPR v, lanes 0-15 -> M=v, lanes 16-31 -> M=v+8; N = lane%16.
  const int nout = col0 + (lane & 15);
  const float bb = bias[nout];
  const int mbase = row0 + ((lane >> 4) << 3);
#pragma unroll
  for (int v = 0; v < 8; ++v) {
    float val = acc[v] + bb;
    if constexpr (RELU) val = fmaxf(val, 0.0f);
    Y[(size_t)(mbase + v) * NCOLS + nout] = val;
  }
}

}  // namespace

extern "C" void kernel_launch(void* const* d_in, const int* in_sizes, int n_in,
                              void* d_out, int out_size, void* d_ws, size_t ws_size,
                              hipStream_t stream) {
  (void)in_sizes; (void)n_in; (void)out_size; (void)ws_size;

  const float* features = (const float*)d_in[0];
  const int*   src      = (const int*)d_in[1];
  const int*   dst      = (const int*)d_in[2];
  const float* weight   = (const float*)d_in[3];
  const float* W1 = (const float*)d_in[4];
  const float* b1 = (const float*)d_in[5];
  const float* W2 = (const float*)d_in[6];
  const float* b2 = (const float*)d_in[7];
  const float* Wo = (const float*)d_in[8];
  const float* bo = (const float*)d_in[9];
  float* out = (float*)d_out;

  float* ws    = (float*)d_ws;
  float* h1    = ws;                       // [NN*FF]
  float* h2    = h1 + (size_t)NN * FF;     // [NN*FF]
  float* summ  = h2 + (size_t)NN * FF;     // [NN*FF]
  float* deg   = summ + (size_t)NN * FF;   // [NN]
  float* sumw  = deg + NN;                 // [NN] (contiguous after deg)
  float* alpha = sumw + NN;                // [NN]
  float* beta  = alpha + NN;               // [NN]

  // Graph statistics (shared by both layers).
  fill_zero_kernel<<<(2 * NN + 255) / 256, 256, 0, stream>>>(deg, 2 * NN);
  edge_stats_kernel<<<NE / 256, 256, 0, stream>>>(dst, weight, deg, sumw);
  node_coef_kernel<<<(NN + 255) / 256, 256, 0, stream>>>(deg, sumw, alpha, beta);

  const int scatterBlocks = (NE * 32) / 256;  // one wave per edge
  const int fillBlocks = (NN * FF) / 256;

  // Layer 1: aggregate features, then fused neigh + 128x128 GEMM + ReLU.
  fill_zero_kernel<<<fillBlocks, 256, 0, stream>>>(summ, NN * FF);
  scatter_kernel<<<scatterBlocks, 256, 0, stream>>>(features, src, dst, weight, summ);
  gemm_wmma_kernel<FF, true, true, 1, 8><<<NN / 16, 256, 0, stream>>>(
      summ, features, alpha, beta, W1, b1, h1);

  // Layer 2.
  fill_zero_kernel<<<fillBlocks, 256, 0, stream>>>(summ, NN * FF);
  scatter_kernel<<<scatterBlocks, 256, 0, stream>>>(h1, src, dst, weight, summ);
  gemm_wmma_kernel<FF, true, true, 1, 8><<<NN / 16, 256, 0, stream>>>(
      summ, h1, alpha, beta, W2, b2, h2);

  // Classifier: 128 -> 32, no activation, no neigh fuse.
  gemm_wmma_kernel<CC, false, false, 4, 2><<<NN / 64, 256, 0, stream>>>(
      h2, h2, nullptr, nullptr, Wo, bo, out);
}